// NeuralDecisionTree_10230612099220
// MI455X (gfx1250) — compile-verified
//
#include <hip/hip_runtime.h>
#include <hip/hip_bf16.h>

// Problem constants (from reference): L=2, B=16, T=512, H=12, NUM_CUT=1,
// NUM_LEAF=2^12=4096, NUM_CLASSES=1000.
#define B_DIM   16
#define T_DIM   512
#define H_DIM   12
#define NLEAF   4096
#define NCLS    1000
#define T_CHUNK 32
#define CHUNKS  (T_DIM / T_CHUNK)   // 16
#define KSPLIT  4                   // waves per GEMM tile (K partitioning)

typedef float v2f __attribute__((ext_vector_type(2)));
typedef float v8f __attribute__((ext_vector_type(8)));

// ---------------------------------------------------------------------------
// Kernel 0: zero the S accumulator (16 x 4096 floats) in workspace.
// ---------------------------------------------------------------------------
__global__ __launch_bounds__(256) void ndt_zero(float* __restrict__ S, int n) {
    int i = blockIdx.x * 256 + threadIdx.x;
    if (i < n) S[i] = 0.0f;
}

// ---------------------------------------------------------------------------
// Kernel 1: S[b][l] = sum_t prod_i h[b,t,i,bit_{11-i}(l)]
//   h[...,i,0] = x ; h[...,i,1] = 2x - cut_i
// Grid: 16 batches x 16 T-chunks. Block: 256 threads.
// Thread tid owns the 16 leaves l = tid*16 + j  (tid = top 8 bits of l,
// j = bottom 4 bits -> features 8..11).
// ---------------------------------------------------------------------------
__global__ __launch_bounds__(256) void ndt_leaf_reduce(
    const float* __restrict__ xl,    // x[-1]: (B, T, H) row-major
    const float* __restrict__ cuts,  // (H, 1)
    float* __restrict__ S)           // (B, NLEAF)
{
    const int b     = blockIdx.x / CHUNKS;
    const int chunk = blockIdx.x % CHUNKS;
    const int tid   = threadIdx.x;

    __shared__ float xs[T_CHUNK * H_DIM];  // 384 floats
    __shared__ float cs[H_DIM];

    const float* xb = xl + ((size_t)b * T_DIM + (size_t)chunk * T_CHUNK) * H_DIM;
    for (int i = tid; i < T_CHUNK * H_DIM; i += 256) xs[i] = xb[i];
    if (tid < H_DIM) cs[tid] = cuts[tid];
    __syncthreads();

    float acc[16];
#pragma unroll
    for (int j = 0; j < 16; ++j) acc[j] = 0.0f;

    for (int t = 0; t < T_CHUNK; ++t) {
        const float* xt = &xs[t * H_DIM];

        // base = product over features 0..7, choice = bit (7-i) of tid
        float base = 1.0f;
#pragma unroll
        for (int i = 0; i < 8; ++i) {
            float xv = xt[i];
            float h1 = 2.0f * xv - cs[i];
            base *= (((tid >> (7 - i)) & 1) ? h1 : xv);
        }

        // tail tree over features 8..11 (j bit3 -> f8, ... , bit0 -> f11)
        float h8a = xt[8],  h8b = 2.0f * xt[8]  - cs[8];
        float h9a = xt[9],  h9b = 2.0f * xt[9]  - cs[9];
        float hAa = xt[10], hAb = 2.0f * xt[10] - cs[10];
        float hBa = xt[11], hBb = 2.0f * xt[11] - cs[11];
        float p2[2] = { h8a, h8b };
        float p4[4], p8[8], p16[16];
#pragma unroll
        for (int j = 0; j < 4; ++j)  p4[j]  = p2[j >> 1] * ((j & 1) ? h9b : h9a);
#pragma unroll
        for (int j = 0; j < 8; ++j)  p8[j]  = p4[j >> 1] * ((j & 1) ? hAb : hAa);
#pragma unroll
        for (int j = 0; j < 16; ++j) p16[j] = p8[j >> 1] * ((j & 1) ? hBb : hBa);
#pragma unroll
        for (int j = 0; j < 16; ++j) acc[j] = fmaf(base, p16[j], acc[j]);
    }

    float* Sb = S + (size_t)b * NLEAF + (size_t)tid * 16;
#pragma unroll
    for (int j = 0; j < 16; ++j) atomicAdd(&Sb[j], acc[j]);  // global_atomic_add_f32
}

// ---------------------------------------------------------------------------
// Kernel 2: out[b][c] = (1/T) * sum_l S[b][l] * score[l][c]
// M=16 (batch), K=4096, N=1000, fp32 WMMA (V_WMMA_F32_16X16X4_F32).
// Block = 4 waves; each wave handles K/4 = 1024 of the reduction for the same
// 16-column tile, then wave 0 reduces the 4 partial C fragments through LDS.
//
// Fragment layouts (ISA 7.12.2):
//   A (16x4): lanes 0-15  -> {A[m][k+0], A[m][k+1]}  (m = lane)
//             lanes 16-31 -> {A[m][k+2], A[m][k+3]}  (m = lane-16)
//   B (4x16): v0: lanes 0-15 = B[k+0][n], lanes 16-31 = B[k+2][n]
//             v1: lanes 0-15 = B[k+1][n], lanes 16-31 = B[k+3][n]
//   C/D (16x16): VGPR r: lanes 0-15 -> M=r, N=lane; lanes 16-31 -> M=r+8.
// ---------------------------------------------------------------------------
__global__ __launch_bounds__(32 * KSPLIT) void ndt_gemm_wmma(
    const float* __restrict__ S,      // (16, 4096)   A, row-major
    const float* __restrict__ score,  // (4096, 1000) B, row-major
    float* __restrict__ out)          // (16, 1000)
{
    const int wid  = threadIdx.x >> 5;   // 0..3
    const int lane = threadIdx.x & 31;   // 0..31
    const int half = lane >> 4;          // 0: K pair {0,1}; 1: K pair {2,3}
    const int l16  = lane & 15;
    const int n0   = blockIdx.x * 16;

    const int kBeg = wid * (NLEAF / KSPLIT);
    const int kEnd = kBeg + (NLEAF / KSPLIT);

    v8f c = {0.f, 0.f, 0.f, 0.f, 0.f, 0.f, 0.f, 0.f};

    const float* arow = S + (size_t)l16 * NLEAF + half * 2;  // + k per iter

    if (n0 + 16 <= NCLS) {
        // ---- fast path: all 16 columns in range, fully branch-free ----
        const float* brow = score + (size_t)(kBeg + half * 2) * NCLS + (n0 + l16);
        for (int k = kBeg; k < kEnd; k += 4) {
            v2f a;
            a.x = arow[k];
            a.y = arow[k + 1];
            v2f bf;
            bf.x = brow[0];
            bf.y = brow[NCLS];
            brow += 4 * NCLS;
            c = __builtin_amdgcn_wmma_f32_16x16x4_f32(
                    false, a, false, bf, (short)0, c, false, false);
        }
    } else {
        // ---- tail tile: clamp addresses in-bounds, zero via multiplier ----
        const int   colB = n0 + l16;
        const float msk  = (colB < NCLS) ? 1.0f : 0.0f;
        const int   colA = (colB < NCLS) ? colB : (NCLS - 1);
        const float* brow = score + (size_t)(kBeg + half * 2) * NCLS + colA;
        for (int k = kBeg; k < kEnd; k += 4) {
            v2f a;
            a.x = arow[k];
            a.y = arow[k + 1];
            v2f bf;
            bf.x = brow[0]    * msk;
            bf.y = brow[NCLS] * msk;
            brow += 4 * NCLS;
            c = __builtin_amdgcn_wmma_f32_16x16x4_f32(
                    false, a, false, bf, (short)0, c, false, false);
        }
    }

    // ---- reduce the KSPLIT partial C fragments through LDS ----
    __shared__ float red[KSPLIT][32][8];   // 4 KB
#pragma unroll
    for (int r = 0; r < 8; ++r) red[wid][lane][r] = c[r];
    __syncthreads();

    if (wid == 0) {
        const int col = n0 + l16;
        if (col < NCLS) {
            const float scale = 1.0f / (float)T_DIM;
#pragma unroll
            for (int r = 0; r < 8; ++r) {
                float v = red[0][lane][r] + red[1][lane][r] +
                          red[2][lane][r] + red[3][lane][r];
                int m = r + half * 8;
                out[(size_t)m * NCLS + col] = v * scale;
            }
        }
    }
}

// ---------------------------------------------------------------------------
extern "C" void kernel_launch(void* const* d_in, const int* in_sizes, int n_in,
                              void* d_out, int out_size, void* d_ws, size_t ws_size,
                              hipStream_t stream) {
    const float* x     = (const float*)d_in[0];  // (L, 16, 512, 12)
    const float* cuts  = (const float*)d_in[1];  // (12, 1)
    const float* score = (const float*)d_in[2];  // (4096, 1000)
    float*       out   = (float*)d_out;          // (16, 1000)
    float*       S     = (float*)d_ws;           // 16*4096 floats = 256 KB

    // x[-1] : last layer, robust to L via in_sizes
    const float* xl = x + (in_sizes[0] - B_DIM * T_DIM * H_DIM);

    const int n_s = B_DIM * NLEAF;
    ndt_zero<<<(n_s + 255) / 256, 256, 0, stream>>>(S, n_s);
    ndt_leaf_reduce<<<B_DIM * CHUNKS, 256, 0, stream>>>(xl, cuts, S);
    ndt_gemm_wmma<<<(NCLS + 15) / 16, 32 * KSPLIT, 0, stream>>>(S, score, out);
}